// VectorizedQuantumCircuit_82222853915149
// MI455X (gfx1250) — compile-verified
//
#include <hip/hip_runtime.h>
#include <hip/hip_bf16.h>

#define NQ 8
#define DIM 256
#define NL 3
#define BATCH 16384
#define MT 32          // M-tile rows per block
#define LDK 260        // padded LDS stride (floats) to spread banks

typedef __attribute__((ext_vector_type(2))) float v2f;
typedef __attribute__((ext_vector_type(4))) float v4f;
typedef __attribute__((ext_vector_type(8))) float v8f;

// ---------------------------------------------------------------------------
// Kernel 1: build the fixed 256x256 unitary U (3 layers of RY/CNOT/RZ).
// Block c evolves basis state |c>; 128 threads share a 256-amplitude LDS state.
// Output layout (for single-b128 B-fragment loads in the GEMM):
//   utp[(k>>1)*1024 + n*4 + 0] = Re U[n][k]     (k even in pair)
//   utp[(k>>1)*1024 + n*4 + 1] = Re U[n][k+1]
//   utp[(k>>1)*1024 + n*4 + 2] = Im U[n][k]
//   utp[(k>>1)*1024 + n*4 + 3] = Im U[n][k+1]
// Block c produces column c of U, i.e. U[j][c] for all j -> k = c, n = j.
// ---------------------------------------------------------------------------
__global__ __launch_bounds__(128) void build_unitary(
    const float* __restrict__ theta, const float* __restrict__ omega,
    const float* __restrict__ attn, float* __restrict__ utp)
{
    __shared__ float2 amp[DIM];
    __shared__ int targ[NQ];
    const int tid = threadIdx.x;
    const int col = blockIdx.x;

    for (int j = tid; j < DIM; j += 128)
        amp[j] = make_float2(j == col ? 1.0f : 0.0f, 0.0f);

    if (tid < NQ) {                      // row-wise argmax (softmax monotonic)
        int best = 0; float bv = attn[tid * NQ];
        for (int j = 1; j < NQ; ++j) {
            float v = attn[tid * NQ + j];
            if (v > bv) { bv = v; best = j; }   // first-max on ties, like jnp
        }
        targ[tid] = best;
    }
    __syncthreads();

    for (int l = 0; l < NL; ++l) {
        // ---- RY(theta[l][q]) on qubit q (bit position 7-q, MSB ordering) ----
        for (int q = 0; q < NQ; ++q) {
            float a = 0.5f * theta[l * NQ + q];
            float c = cosf(a), s = sinf(a);
            int p = 7 - q;
            int mask = (1 << p) - 1;
            {   // 128 pairs, one per thread
                int idx0 = ((tid & ~mask) << 1) | (tid & mask);
                int idx1 = idx0 | (1 << p);
                float2 a0 = amp[idx0], a1 = amp[idx1];
                amp[idx0] = make_float2(c * a0.x - s * a1.x, c * a0.y - s * a1.y);
                amp[idx1] = make_float2(s * a0.x + c * a1.x, s * a0.y + c * a1.y);
            }
            __syncthreads();
        }
        // ---- CNOT(control q, target targ[q]); identity if equal ----
        for (int q = 0; q < NQ; ++q) {
            int t = targ[q];
            if (t != q) {
                int cp = 7 - q, tp = 7 - t;
                if (tid < 64) {  // 64 swap pairs: control bit = 1, target bit enumerated
                    int idx = 0, bp = 0;
                    #pragma unroll
                    for (int b = 0; b < 8; ++b) {
                        if (b == cp || b == tp) continue;
                        idx |= ((tid >> bp) & 1) << b; ++bp;
                    }
                    idx |= (1 << cp);
                    int idx2 = idx | (1 << tp);
                    float2 u = amp[idx], v = amp[idx2];
                    amp[idx] = v; amp[idx2] = u;
                }
            }
            __syncthreads();
        }
        // ---- RZ(omega[l][q]): phase -w/2 on bit=0, +w/2 on bit=1 ----
        for (int q = 0; q < NQ; ++q) {
            float h = 0.5f * omega[l * NQ + q];
            float ch = cosf(h), sh = sinf(h);
            int p = 7 - q;
            for (int j = tid; j < DIM; j += 128) {
                float sn = (((j >> p) & 1) ? sh : -sh);
                float2 a = amp[j];
                amp[j] = make_float2(a.x * ch - a.y * sn, a.x * sn + a.y * ch);
            }
            __syncthreads();
        }
    }
    // Write column c into the k-pair-interleaved packed layout.
    {
        const int pair = col >> 1;
        const int po   = col & 1;
        float* base = utp + (size_t)pair * 1024;
        for (int j = tid; j < DIM; j += 128) {
            base[j * 4 + 0 + po] = amp[j].x;   // re
            base[j * 4 + 2 + po] = amp[j].y;   // im
        }
    }
}

// ---------------------------------------------------------------------------
// Kernel 2: out[b, n] = sum_k Psi[b, k] * U[n][k]  (complex, Psi real)
// Psi built on the fly in LDS from x (product-state encoding).
// Each block: 32 rows x 256 cols (complex). 8 waves, each a 32-col strip,
// 2x2x{re,im} = 8 accumulators of v_wmma_f32_16x16x4_f32.
// B fragments (re+im for one n-subtile, both k-rows) arrive in ONE b128 load.
// ---------------------------------------------------------------------------
__global__ __launch_bounds__(256) void qgemm(
    const float* __restrict__ x,
    const float* __restrict__ utp,
    float* __restrict__ out)
{
    __shared__ float psi[MT * LDK];
    const int tid = threadIdx.x;
    const int row0 = blockIdx.x * MT;

    // ---- Phase 1: Psi tile (32 x 256) in LDS ----
    {
        int r  = tid >> 3;   // 0..31 row in tile
        int kg = tid & 7;    // 0..7 -> owns 32 k-values (top 3 bits of k fixed)
        const float* xr = x + (size_t)(row0 + r) * NQ;
        float c[NQ], s[NQ];
        #pragma unroll
        for (int i = 0; i < NQ; ++i) {
            float a = 0.5f * xr[i];
            c[i] = cosf(a); s[i] = sinf(a);
        }
        // qubit i <-> bit (7-i). Top bits 7,6,5 = qubits 0,1,2 come from kg.
        float base = (((kg >> 2) & 1) ? s[0] : c[0]) *
                     (((kg >> 1) & 1) ? s[1] : c[1]) *
                     (((kg >> 0) & 1) ? s[2] : c[2]);
        #pragma unroll
        for (int lo = 0; lo < 32; ++lo) {
            float p = base;
            #pragma unroll
            for (int i = 3; i < 8; ++i)
                p *= ((lo >> (7 - i)) & 1) ? s[i] : c[i];
            psi[r * LDK + kg * 32 + lo] = p;
        }
    }
    __syncthreads();

    // ---- Phase 2: WMMA GEMM ----
    const int wave  = tid >> 5;
    const int lane  = tid & 31;
    const int ml    = lane & 15;           // row/col within 16
    const int khalf = (lane >> 4) << 1;    // lanes 16-31 hold K+2,K+3
    const int nbase = wave * 32;

    v8f acc[2][2][2] = {};                 // [msub][nsub][re/im]

    for (int k = 0; k < DIM; k += 4) {
        int kk = k + khalf;
        // A fragments: 16x4 f32, A[m][kk..kk+1] from LDS
        v2f afr[2];
        #pragma unroll
        for (int ms = 0; ms < 2; ++ms) {
            const float* p = &psi[(ms * 16 + ml) * LDK + kk];
            v2f a; a.x = p[0]; a.y = p[1];
            afr[ms] = a;
        }
        // B fragments: one b128 per n-subtile carries re[kk], re[kk+1], im[kk], im[kk+1]
        v2f bre[2], bim[2];
        #pragma unroll
        for (int ns = 0; ns < 2; ++ns) {
            int n = nbase + ns * 16 + ml;
            const v4f b4 = *(const v4f*)(utp + ((size_t)(kk >> 1) * 1024 + n * 4));
            v2f br; br.x = b4.x; br.y = b4.y;
            v2f bi; bi.x = b4.z; bi.y = b4.w;
            bre[ns] = br; bim[ns] = bi;
        }
        #pragma unroll
        for (int ms = 0; ms < 2; ++ms)
            #pragma unroll
            for (int ns = 0; ns < 2; ++ns) {
                acc[ms][ns][0] = __builtin_amdgcn_wmma_f32_16x16x4_f32(
                    false, afr[ms], false, bre[ns], (short)0, acc[ms][ns][0], false, false);
                acc[ms][ns][1] = __builtin_amdgcn_wmma_f32_16x16x4_f32(
                    false, afr[ms], false, bim[ns], (short)0, acc[ms][ns][1], false, false);
            }
    }

    // ---- Store: D 16x16 f32 layout: VGPR v -> M = v (+8 for lanes 16-31), N = lane&15
    const int nloc = lane & 15;
    const int mhi  = (lane >> 4) << 3;
    float2* out2 = (float2*)out;
    #pragma unroll
    for (int ms = 0; ms < 2; ++ms)
        #pragma unroll
        for (int ns = 0; ns < 2; ++ns)
            #pragma unroll
            for (int v = 0; v < 8; ++v) {
                int m = row0 + ms * 16 + mhi + v;
                int n = nbase + ns * 16 + nloc;
                float2 o;
                o.x = acc[ms][ns][0][v];
                o.y = acc[ms][ns][1][v];
                out2[(size_t)m * DIM + n] = o;
            }
}

// ---------------------------------------------------------------------------
extern "C" void kernel_launch(void* const* d_in, const int* in_sizes, int n_in,
                              void* d_out, int out_size, void* d_ws, size_t ws_size,
                              hipStream_t stream) {
    const float* x     = (const float*)d_in[0];   // (16384, 8)
    const float* theta = (const float*)d_in[1];   // (3, 8)
    const float* omega = (const float*)d_in[2];   // (3, 8)
    const float* attn  = (const float*)d_in[3];   // (8, 8)

    float* utp = (float*)d_ws;                    // 128 * 1024 f32 = 512 KB packed U^T

    build_unitary<<<DIM, 128, 0, stream>>>(theta, omega, attn, utp);
    qgemm<<<BATCH / MT, 256, 0, stream>>>(x, utp, (float*)d_out);
}